// AttentionLayer_5196910428562
// MI455X (gfx1250) — compile-verified
//
#include <hip/hip_runtime.h>
#include <math.h>

// ---------------- problem constants (match reference) ----------------
#define B_SZ     1024
#define D_IN     128
#define H_NUM    8
#define N_HIST   200
#define N_PAD    208          // 13 tiles of 16
#define N_TILES  13
#define RS       132          // padded LDS row stride in floats (132 % 64banks = 4 -> conflict free)

typedef __attribute__((ext_vector_type(2))) float v2f;
typedef __attribute__((ext_vector_type(8))) float v8f;
typedef __attribute__((ext_vector_type(4))) unsigned int u32x4;
typedef __attribute__((ext_vector_type(4))) int i32x4;
typedef __attribute__((ext_vector_type(8))) int i32x8;

// =====================================================================
// TDM: one-instruction DMA of a 2D f32 tile (dim0 x dim1, row stride = dim0
// elements) from global memory into LDS, inserting 4 DWORDs of padding after
// every 128 DWORDs so rows land at a 132-float stride (bank-conflict free).
//   D# group0: count=1 | lds_addr | global_addr | type=2
//   D# group1: data_size=4B, pad_enable, pad_interval=128dw, pad_amount=4dw,
//              tensor_dim0/1, tile_dim0/1, tensor_dim0_stride
// =====================================================================
__device__ __forceinline__ void tdm_load_2d_padded(unsigned lds_addr,
                                                   const float* gptr,
                                                   int dim0, int dim1) {
  unsigned long long ga = (unsigned long long)(uintptr_t)gptr;
  u32x4 g0 = { 1u,                                   // count=1, user descriptor
               lds_addr,                              // LDS byte address
               (unsigned)(ga & 0xFFFFFFFFu),          // global_addr[31:0]
               ((unsigned)((ga >> 32) & 0x01FFFFFFu)) // global_addr[56:32]
                 | 0x80000000u };                     // type=2 ("image")
  i32x8 g1 = { (int)0x07920000,   // data_size=2(4B) | pad_en | interval=6(128dw) | amount=3(4dw)
               dim0 << 16,        // [63:48]  tensor_dim0[15:0]
               dim1 << 16,        // [79:64]  tensor_dim0[31:16]=0 | [95:80] tensor_dim1[15:0]
               dim0 << 16,        // [111:96] tensor_dim1[31:16]=0 | [127:112] tile_dim0
               dim1,              // [143:128] tile_dim1 | [159:144] tile_dim2=0
               dim0,              // [191:160] tensor_dim0_stride[31:0]
               0,                 // stride0[47:32]=0 | stride1[15:0]=0
               0 };               // stride1[47:16]=0
  i32x4 z4 = {0, 0, 0, 0};
#if __clang_major__ >= 23
  i32x8 z8 = {0, 0, 0, 0, 0, 0, 0, 0};
  __builtin_amdgcn_tensor_load_to_lds(g0, g1, z4, z4, z8, 0);
#else
  __builtin_amdgcn_tensor_load_to_lds(g0, g1, z4, z4, 0);
#endif
}

// =====================================================================
// Kernel 1: pq[b, j] = tanh( sum_d q[b,d] * Wq[d,j] + bq[j] ),  j in [0,1024)
// Tiled 64 batches x 256 cols per block so Wq L2 traffic stays ~8 MB total.
// =====================================================================
__global__ __launch_bounds__(256) void proj_query_kernel(
    const float* __restrict__ q, const float* __restrict__ Wq,
    const float* __restrict__ bq, float* __restrict__ pq) {
  __shared__ float qs[64][D_IN];               // 32 KB
  const int tid = threadIdx.x;
  const int j   = blockIdx.y * 256 + tid;      // output column
  const int b0  = blockIdx.x * 64;             // batch tile base

  for (int i = tid; i < 64 * D_IN; i += 256)
    qs[i >> 7][i & 127] = q[(size_t)(b0 + (i >> 7)) * D_IN + (i & 127)];
  __syncthreads();

  float acc[64];
#pragma unroll
  for (int bb = 0; bb < 64; ++bb) acc[bb] = 0.f;

  for (int d = 0; d < D_IN; ++d) {
    const float w = Wq[(size_t)d * 1024 + j];  // coalesced across lanes
#pragma unroll
    for (int bb = 0; bb < 64; ++bb) acc[bb] = fmaf(qs[bb][d], w, acc[bb]);
  }

  const float bias = bq[j];
  for (int bb = 0; bb < 64; ++bb)
    pq[(size_t)(b0 + bb) * 1024 + j] = tanhf(acc[bb] + bias);
}

// =====================================================================
// Kernel 2: one workgroup (256 thr = 8 wave32) per batch.
//  A) TDM-stage hist[b] (200x128 f32) + pq[b] (8x128) into LDS with
//     hardware-inserted padding; zero-fill pad rows with ds stores
//  B) dot[n,h] via V_WMMA_F32_16X16X4_F32 on 16x16 tiles, K=128
//     + Wr reduction via xor-shuffles -> scores[n] (+ additive mask)
//  C) block softmax over n<200
//  D) out[d] = (1/sum) * sum_n w[n] * hist[n,d]  (LDS resident)
// =====================================================================
__global__ __launch_bounds__(256, 2) void attn_fused_kernel(
    const float* __restrict__ hist, const float* __restrict__ mask,
    const float* __restrict__ pq,   const float* __restrict__ Wr,
    float* __restrict__ out) {
  __shared__ __align__(16) float hs[N_PAD * RS];   // 109,824 B (hist, padded stride)
  __shared__ __align__(16) float ps[16 * RS];      //   8,448 B (pq rows 0-7, zeros 8-15)
  __shared__ float sc[N_PAD];          // scores / softmax weights
  __shared__ float red[8];             // cross-wave reduction
  __shared__ float part[2 * D_IN];     // phase-D partial sums

  const int tid  = threadIdx.x;
  const int lane = tid & 31;
  const int wv   = tid >> 5;           // wave id 0..7
  const int b    = blockIdx.x;

  // ---------------- A) stage to LDS via Tensor Data Mover ----------------
  // zero-fill the pad rows (disjoint from TDM destinations)
  for (int i = tid; i < (N_PAD - N_HIST) * RS; i += 256)
    hs[N_HIST * RS + i] = 0.f;
  for (int i = tid; i < H_NUM * RS; i += 256)
    ps[H_NUM * RS + i] = 0.f;

  if (wv == 0) {
    tdm_load_2d_padded((unsigned)(uintptr_t)&hs[0],
                       hist + (size_t)b * N_HIST * D_IN, D_IN, N_HIST);
    __builtin_amdgcn_s_wait_tensorcnt(0);
  } else if (wv == 1) {
    tdm_load_2d_padded((unsigned)(uintptr_t)&ps[0],
                       pq + (size_t)b * 1024, D_IN, H_NUM);
    __builtin_amdgcn_s_wait_tensorcnt(0);
  }
  __syncthreads();

  // ---------------- B) WMMA dot + Wr reduction ----------------
  {
    const int m    = lane & 15;          // N-column of D (== head id), and M-row index for A frag
    const int koff = (lane >> 4) << 1;   // K sub-offset per A/B fragment layout
    const float wr = (m < H_NUM) ? Wr[m] : 0.f;

    for (int T = wv; T < N_TILES; T += 8) {
      v8f c = {0.f, 0.f, 0.f, 0.f, 0.f, 0.f, 0.f, 0.f};
      const int abase = (T * 16 + m) * RS + koff;
      const int bbase = m * RS + koff;
#pragma unroll 4
      for (int k0 = 0; k0 < D_IN; k0 += 4) {
        v2f a  = *reinterpret_cast<const v2f*>(&hs[abase + k0]);   // ds_load_b64, conflict-free
        v2f bm = *reinterpret_cast<const v2f*>(&ps[bbase + k0]);
        c = __builtin_amdgcn_wmma_f32_16x16x4_f32(
            /*neg_a=*/false, a, /*neg_b=*/false, bm,
            /*c_mod=*/(short)0, c, /*reuse_a=*/false, /*reuse_b=*/false);
      }
      // D layout: lanes 0-15 hold M = vgpr, lanes 16-31 hold M = vgpr+8; N = lane&15.
      // scores[M] = sum_h Wr[h] * D[M, h]  -> scale by Wr[N] then xor-reduce across 16 lanes.
#pragma unroll
      for (int j = 0; j < 8; ++j) {
        float v = c[j] * wr;
        v += __shfl_xor(v, 1, 32);
        v += __shfl_xor(v, 2, 32);
        v += __shfl_xor(v, 4, 32);
        v += __shfl_xor(v, 8, 32);
        const int row = T * 16 + j + ((lane >= 16) ? 8 : 0);
        if (m == 0 && row < N_HIST)
          sc[row] = v + mask[(size_t)b * N_HIST + row];
      }
    }
  }
  __syncthreads();

  // ---------------- C) softmax over n<200 ----------------
  float lmax = -INFINITY;
  for (int n = tid; n < N_HIST; n += 256) lmax = fmaxf(lmax, sc[n]);
#pragma unroll
  for (int off = 16; off > 0; off >>= 1) lmax = fmaxf(lmax, __shfl_xor(lmax, off, 32));
  if (lane == 0) red[wv] = lmax;
  __syncthreads();
  float gmax = red[0];
#pragma unroll
  for (int i = 1; i < 8; ++i) gmax = fmaxf(gmax, red[i]);
  __syncthreads();

  float lsum = 0.f;
  for (int n = tid; n < N_HIST; n += 256) {
    const float e = __expf(sc[n] - gmax);
    sc[n] = e;
    lsum += e;
  }
#pragma unroll
  for (int off = 16; off > 0; off >>= 1) lsum += __shfl_xor(lsum, off, 32);
  if (lane == 0) red[wv] = lsum;
  __syncthreads();
  float gsum = 0.f;
#pragma unroll
  for (int i = 0; i < 8; ++i) gsum += red[i];
  const float inv = 1.f / gsum;
  __syncthreads();

  // ---------------- D) weighted sum over history ----------------
  {
    const int d    = tid & 127;
    const int half = tid >> 7;                 // 0 or 1 -> n ranges [0,100) / [100,200)
    float acc = 0.f;
    const int n0 = half * 100;
    for (int n = n0; n < n0 + 100; ++n)
      acc = fmaf(sc[n], hs[n * RS + d], acc);  // lanes hit distinct banks
    part[half * D_IN + d] = acc;
  }
  __syncthreads();
  if (tid < D_IN)
    out[(size_t)b * D_IN + tid] = (part[tid] + part[D_IN + tid]) * inv;
}

// =====================================================================
extern "C" void kernel_launch(void* const* d_in, const int* in_sizes, int n_in,
                              void* d_out, int out_size, void* d_ws, size_t ws_size,
                              hipStream_t stream) {
  const float* hist  = (const float*)d_in[0];  // [1024, 200, 128]
  const float* mask  = (const float*)d_in[1];  // [1024, 200]
  const float* query = (const float*)d_in[2];  // [1024, 128]
  const float* Wq    = (const float*)d_in[3];  // [128, 1024]
  const float* bq    = (const float*)d_in[4];  // [1024]
  const float* Wr    = (const float*)d_in[5];  // [8]
  float* out = (float*)d_out;                  // [1024, 128]
  float* pq  = (float*)d_ws;                   // [1024, 1024] = 4 MB scratch

  dim3 g1(B_SZ / 64, 1024 / 256);
  proj_query_kernel<<<g1, 256, 0, stream>>>(query, Wq, bq, pq);
  attn_fused_kernel<<<B_SZ, 256, 0, stream>>>(hist, mask, pq, Wr, out);
}